// MambaBlock_16990890622993
// MI455X (gfx1250) — compile-verified
//
#include <hip/hip_runtime.h>

typedef __attribute__((ext_vector_type(2))) float v2f;
typedef __attribute__((ext_vector_type(8))) float v8f;

#define D_MODEL 512
#define SEQ     32768
#define DT_RANK 32
#define D_STATE 2
#define D_CONV  4
#define CHUNK   128
#define NCHUNK  (SEQ / CHUNK)   // 256
#define KT      32              // GEMM K-tile

// ---------------------------------------------------------------------------
// CDNA5 async global -> LDS copy (16B per lane), tracked by ASYNCcnt.
// lds_addr = LDS byte address (flat shared address truncated to 32 bits,
// per ISA aperture rule).
// ---------------------------------------------------------------------------
__device__ __forceinline__ void async_b128(uint32_t lds_addr, const float* gptr) {
    asm volatile("global_load_async_to_lds_b128 %0, %1, off"
                 :: "v"(lds_addr), "v"(gptr) : "memory");
}
__device__ __forceinline__ void wait_async0() {
    asm volatile("s_wait_asynccnt 0x0" ::: "memory");
}

// ---------------------------------------------------------------------------
// Epilogues: 0 none | 1 softplus(v + bias[col]) | 2 finite-clamp
// ---------------------------------------------------------------------------
template <int EPI>
__device__ __forceinline__ float apply_epilogue(float v, const float* bias, int col) {
    if (EPI == 1) {
        v += bias[col];
        v = (v > 20.0f) ? v : log1pf(expf(v));
    } else if (EPI == 2) {
        v = __builtin_isfinite(v) ? v : 0.0f;
    }
    return v;
}

// ---------------------------------------------------------------------------
// fp32 WMMA GEMM:  C[M,N] = A[M,K] * B      (TRANS_B=false: B is K x N row-major)
//                  C[M,N] = A[M,K] * W^T    (TRANS_B=true : W is N x K row-major)
// Requires M%64==0, N%64==0, K%KT==0.
// Block: 128 threads (4 waves). Block tile 64x64, KT=32, double-buffered LDS
// filled with GLOBAL_LOAD_ASYNC_TO_LDS_B128 (ASYNCcnt), V_WMMA_F32_16X16X4_F32
// compute. Each wave owns a 32x32 tile (2x2 accumulators).
// ---------------------------------------------------------------------------
template <bool TRANS_B, int EPI>
__global__ __launch_bounds__(128) void gemm_wmma_f32(
    const float* __restrict__ A, const float* __restrict__ B,
    const float* __restrict__ bias, float* __restrict__ C,
    int M, int N, int K)
{
    constexpr int BROWS = TRANS_B ? 64 : KT;
    constexpr int BCOLS = TRANS_B ? (KT + 4) : (64 + 4);
    __shared__ float sA[2][64][KT + 4];       // M x K tile (row-major)
    __shared__ float sB[2][BROWS][BCOLS];     // N x K if TRANS_B else K x N

    const int tid  = threadIdx.x;
    const int lane = tid & 31;
    const int wave = tid >> 5;
    const int wm   = wave >> 1;        // wave row (0..1)
    const int wn   = wave & 1;         // wave col (0..1)
    const int bm   = blockIdx.y * 64;
    const int bn   = blockIdx.x * 64;

    const int l16  = lane & 15;
    const int half = lane >> 4;        // K sub-pair select per ISA A/B layout
    const int hi8  = half * 8;         // C-store row offset

    v8f acc[2][2];
    const v8f vzero = {0.f, 0.f, 0.f, 0.f, 0.f, 0.f, 0.f, 0.f};
#pragma unroll
    for (int mi = 0; mi < 2; ++mi)
#pragma unroll
        for (int ni = 0; ni < 2; ++ni) acc[mi][ni] = vzero;

    auto stageA = [&](int buf, int k0) {       // 64 x 32 floats, 4 b128 / thread
#pragma unroll
        for (int p = 0; p < 4; ++p) {
            int f = (tid + p * 128) * 4;
            int r = f >> 5, c = f & 31;
            async_b128((uint32_t)(uintptr_t)&sA[buf][r][c],
                       A + (size_t)(bm + r) * K + (k0 + c));
        }
    };
    auto stageB = [&](int buf, int k0) {
        if (TRANS_B) {                         // copy W rows (N x K major)
#pragma unroll
            for (int p = 0; p < 4; ++p) {
                int f = (tid + p * 128) * 4;
                int n = f >> 5, c = f & 31;
                async_b128((uint32_t)(uintptr_t)&sB[buf][n][c],
                           B + (size_t)(bn + n) * K + (k0 + c));
            }
        } else {                               // copy B rows (K x N major)
#pragma unroll
            for (int p = 0; p < 4; ++p) {
                int f = (tid + p * 128) * 4;
                int k = f >> 6, n = f & 63;
                async_b128((uint32_t)(uintptr_t)&sB[buf][k][n],
                           B + (size_t)(k0 + k) * N + (bn + n));
            }
        }
    };

    stageA(0, 0);
    stageB(0, 0);
    wait_async0();
    __syncthreads();

    int buf = 0;
    for (int k0 = 0; k0 < K; k0 += KT) {
        const bool has_next = (k0 + KT) < K;   // uniform
        if (has_next) { stageA(buf ^ 1, k0 + KT); stageB(buf ^ 1, k0 + KT); }

#pragma unroll
        for (int kk = 0; kk < KT; kk += 4) {
            const int kb = kk + half * 2;
            v2f a[2], b[2];
#pragma unroll
            for (int mi = 0; mi < 2; ++mi) {
                int r = wm * 32 + mi * 16 + l16;
                a[mi].x = sA[buf][r][kb];
                a[mi].y = sA[buf][r][kb + 1];
            }
#pragma unroll
            for (int ni = 0; ni < 2; ++ni) {
                int c = wn * 32 + ni * 16 + l16;
                if (TRANS_B) {                 // N x K major: contiguous pair
                    b[ni].x = sB[buf][c][kb];
                    b[ni].y = sB[buf][c][kb + 1];
                } else {                       // K x N major: strided pair
                    b[ni].x = sB[buf][kb][c];
                    b[ni].y = sB[buf][kb + 1][c];
                }
            }
#pragma unroll
            for (int mi = 0; mi < 2; ++mi)
#pragma unroll
                for (int ni = 0; ni < 2; ++ni)
                    acc[mi][ni] = __builtin_amdgcn_wmma_f32_16x16x4_f32(
                        false, a[mi], false, b[ni], (short)0, acc[mi][ni],
                        false, false);
        }
        if (has_next) wait_async0();
        __syncthreads();
        buf ^= 1;
    }

    // C store: VGPR r -> rows r (lanes 0-15) / r+8 (lanes 16-31), col = l16
#pragma unroll
    for (int mi = 0; mi < 2; ++mi) {
#pragma unroll
        for (int ni = 0; ni < 2; ++ni) {
            int col = bn + wn * 32 + ni * 16 + l16;
#pragma unroll
            for (int r = 0; r < 8; ++r) {
                int row = bm + wm * 32 + mi * 16 + r + hi8;
                float v = apply_epilogue<EPI>(acc[mi][ni][r], bias, col);
                C[(size_t)row * N + col] = v;
            }
        }
    }
}

// ---------------------------------------------------------------------------
// Depthwise causal conv (D_CONV=4) + SiLU.  xs = xr[:, 0:512] (row stride 1024)
// ---------------------------------------------------------------------------
__global__ __launch_bounds__(256) void conv_silu_kernel(
    const float* __restrict__ xr, const float* __restrict__ conv_w,
    const float* __restrict__ conv_b, float* __restrict__ u)
{
    int idx = blockIdx.x * 256 + threadIdx.x;
    if (idx >= SEQ * D_MODEL) return;
    int t = idx / D_MODEL;
    int d = idx - t * D_MODEL;
    float acc = conv_b[d];
#pragma unroll
    for (int k = 0; k < D_CONV; ++k) {
        int tt = t - (D_CONV - 1) + k;
        if (tt >= 0)
            acc += conv_w[d * D_CONV + k] * xr[(size_t)tt * (2 * D_MODEL) + d];
    }
    u[idx] = acc / (1.0f + expf(-acc));   // silu
}

// ---------------------------------------------------------------------------
// x_proj: x_dbl[t, 0:36] = u[t,:] @ x_proj_w^T   (N=36, K=512)
// ---------------------------------------------------------------------------
__global__ __launch_bounds__(64) void xproj_kernel(
    const float* __restrict__ u, const float* __restrict__ xpw,
    float* __restrict__ delta_raw, float* __restrict__ Bc, float* __restrict__ Cc)
{
    __shared__ float su[D_MODEL];
    int t = blockIdx.x;
#pragma unroll
    for (int i = threadIdx.x; i < D_MODEL / 4; i += 64)
        ((float4*)su)[i] = ((const float4*)(u + (size_t)t * D_MODEL))[i];
    __syncthreads();

    int j = threadIdx.x;
    if (j < DT_RANK + 2 * D_STATE) {
        const float* w = xpw + (size_t)j * D_MODEL;
        float acc = 0.0f;
#pragma unroll 8
        for (int k = 0; k < D_MODEL; ++k) acc += su[k] * w[k];
        if (j < DT_RANK)                 delta_raw[(size_t)t * DT_RANK + j] = acc;
        else if (j < DT_RANK + D_STATE)  Bc[t * 2 + (j - DT_RANK)] = acc;
        else                             Cc[t * 2 + (j - DT_RANK - D_STATE)] = acc;
    }
}

// ---------------------------------------------------------------------------
// Selective scan, chunked 3-pass parallel linear recurrence.
// ---------------------------------------------------------------------------
__global__ __launch_bounds__(D_MODEL) void scan_pass1(
    const float* __restrict__ delta_p, const float* __restrict__ u,
    const float* __restrict__ Bc, const float* __restrict__ A_log,
    float* __restrict__ chunkP, float* __restrict__ chunkH)
{
    __shared__ float sB[CHUNK * 2];
    int c = blockIdx.x;
    int d = threadIdx.x;
    for (int i = d; i < CHUNK * 2; i += D_MODEL) sB[i] = Bc[c * CHUNK * 2 + i];
    __syncthreads();

    float A0 = -expf(A_log[d * 2 + 0]);
    float A1 = -expf(A_log[d * 2 + 1]);
    float P0 = 1.f, P1 = 1.f, h0 = 0.f, h1 = 0.f;
    for (int t = 0; t < CHUNK; ++t) {
        size_t tt = (size_t)c * CHUNK + t;
        float dp = delta_p[tt * D_MODEL + d];
        float uu = u[tt * D_MODEL + d];
        float a0 = expf(dp * A0);
        float a1 = expf(dp * A1);
        float db = dp * uu;
        h0 = a0 * h0 + db * sB[t * 2 + 0];
        h1 = a1 * h1 + db * sB[t * 2 + 1];
        P0 *= a0;
        P1 *= a1;
    }
    int base = c * D_MODEL * 2 + d * 2;
    chunkP[base] = P0; chunkP[base + 1] = P1;
    chunkH[base] = h0; chunkH[base + 1] = h1;
}

__global__ __launch_bounds__(1024) void scan_pass2(
    const float* __restrict__ chunkP, const float* __restrict__ chunkH,
    float* __restrict__ hstart)
{
    int tid = threadIdx.x;          // d*2 + n
    float h = 0.0f;
    for (int c = 0; c < NCHUNK; ++c) {
        int idx = c * (D_MODEL * 2) + tid;
        hstart[idx] = h;
        h = chunkP[idx] * h + chunkH[idx];
    }
}

__global__ __launch_bounds__(D_MODEL) void scan_pass3(
    const float* __restrict__ delta_p, const float* __restrict__ u,
    const float* __restrict__ Bc, const float* __restrict__ Cc,
    const float* __restrict__ A_log, const float* __restrict__ Dv,
    const float* __restrict__ xr, const float* __restrict__ hstart,
    float* __restrict__ y)
{
    __shared__ float sB[CHUNK * 2];
    __shared__ float sC[CHUNK * 2];
    int c = blockIdx.x;
    int d = threadIdx.x;
    for (int i = d; i < CHUNK * 2; i += D_MODEL) {
        sB[i] = Bc[c * CHUNK * 2 + i];
        sC[i] = Cc[c * CHUNK * 2 + i];
    }
    __syncthreads();

    float A0 = -expf(A_log[d * 2 + 0]);
    float A1 = -expf(A_log[d * 2 + 1]);
    float h0 = hstart[c * (D_MODEL * 2) + d * 2 + 0];
    float h1 = hstart[c * (D_MODEL * 2) + d * 2 + 1];
    float Dd = Dv[d];
    for (int t = 0; t < CHUNK; ++t) {
        size_t tt = (size_t)c * CHUNK + t;
        float dp = delta_p[tt * D_MODEL + d];
        float uu = u[tt * D_MODEL + d];
        float a0 = expf(dp * A0);
        float a1 = expf(dp * A1);
        float db = dp * uu;
        h0 = a0 * h0 + db * sB[t * 2 + 0];
        h1 = a1 * h1 + db * sB[t * 2 + 1];
        float ys = h0 * sC[t * 2 + 0] + h1 * sC[t * 2 + 1];
        float r  = xr[tt * (2 * D_MODEL) + D_MODEL + d];       // res
        float sr = r / (1.0f + expf(-r));                      // silu(res)
        y[tt * D_MODEL + d] = (ys + uu * Dd) * sr;
    }
}

// ---------------------------------------------------------------------------
extern "C" void kernel_launch(void* const* d_in, const int* in_sizes, int n_in,
                              void* d_out, int out_size, void* d_ws, size_t ws_size,
                              hipStream_t stream)
{
    const float* x          = (const float*)d_in[0];
    const float* dis_dense  = (const float*)d_in[1];
    const float* in_proj_w  = (const float*)d_in[2];
    const float* conv_w     = (const float*)d_in[3];
    const float* conv_b     = (const float*)d_in[4];
    const float* x_proj_w   = (const float*)d_in[5];
    const float* dt_proj_w  = (const float*)d_in[6];
    const float* dt_proj_b  = (const float*)d_in[7];
    const float* A_log      = (const float*)d_in[8];
    const float* Dv         = (const float*)d_in[9];
    const float* out_proj_w = (const float*)d_in[10];
    float* out = (float*)d_out;

    float* ws = (float*)d_ws;
    size_t off = 0;
    float* xr        = ws + off; off += (size_t)SEQ * 2 * D_MODEL;
    float* u         = ws + off; off += (size_t)SEQ * D_MODEL;
    float* delta_raw = ws + off; off += (size_t)SEQ * DT_RANK;
    float* Bc        = ws + off; off += (size_t)SEQ * 2;
    float* Cc        = ws + off; off += (size_t)SEQ * 2;
    float* delta     = ws + off; off += (size_t)SEQ * D_MODEL;
    float* delta_p   = ws + off; off += (size_t)SEQ * D_MODEL;
    float* chunkP    = ws + off; off += (size_t)NCHUNK * D_MODEL * 2;
    float* chunkH    = ws + off; off += (size_t)NCHUNK * D_MODEL * 2;
    float* hstart    = ws + off; off += (size_t)NCHUNK * D_MODEL * 2;
    float* y         = delta;    // delta dead after delta_p GEMM; reuse

    // 1) in_proj: xr = x @ in_proj_w^T   (M=SEQ, N=1024, K=512)
    gemm_wmma_f32<true, 0><<<dim3((2 * D_MODEL) / 64, SEQ / 64), 128, 0, stream>>>(
        x, in_proj_w, nullptr, xr, SEQ, 2 * D_MODEL, D_MODEL);

    // 2) depthwise causal conv + silu -> u
    conv_silu_kernel<<<(SEQ * D_MODEL + 255) / 256, 256, 0, stream>>>(
        xr, conv_w, conv_b, u);

    // 3) x_proj (N=36): delta_raw, B, C
    xproj_kernel<<<SEQ, 64, 0, stream>>>(u, x_proj_w, delta_raw, Bc, Cc);

    // 4) dt_proj + softplus: delta = softplus(delta_raw @ dt_proj_w^T + b)
    gemm_wmma_f32<true, 1><<<dim3(D_MODEL / 64, SEQ / 64), 128, 0, stream>>>(
        delta_raw, dt_proj_w, dt_proj_b, delta, SEQ, D_MODEL, DT_RANK);

    // 5) delta_p = delta @ dis_dense   (B is K x N row-major)
    gemm_wmma_f32<false, 0><<<dim3(D_MODEL / 64, SEQ / 64), 128, 0, stream>>>(
        delta, dis_dense, nullptr, delta_p, SEQ, D_MODEL, D_MODEL);

    // 6-8) chunked selective scan
    scan_pass1<<<NCHUNK, D_MODEL, 0, stream>>>(delta_p, u, Bc, A_log, chunkP, chunkH);
    scan_pass2<<<1, D_MODEL * 2, 0, stream>>>(chunkP, chunkH, hstart);
    scan_pass3<<<NCHUNK, D_MODEL, 0, stream>>>(delta_p, u, Bc, Cc, A_log, Dv, xr,
                                               hstart, y);

    // 9) out_proj + finite-clamp: out = y @ out_proj_w^T
    gemm_wmma_f32<true, 2><<<dim3(D_MODEL / 64, SEQ / 64), 128, 0, stream>>>(
        y, out_proj_w, nullptr, out, SEQ, D_MODEL, D_MODEL);

    (void)in_sizes; (void)n_in; (void)out_size; (void)ws_size;
}